// ValueNetwork_68453188764138
// MI455X (gfx1250) — compile-verified
//
#include <hip/hip_runtime.h>
#include <hip/hip_bf16.h>

typedef __attribute__((ext_vector_type(2))) float v2f;
typedef __attribute__((ext_vector_type(8))) float v8f;

#define NB      16384
#define NH      64
#define HS_LD   12   // staged human_states (64 x 8 used), conflict-free pairs
#define A1_LD   68   // hidden 64x64 activation stride
#define H_LD    34   // h matrices (64 x 32) stride
#define W1_LD   80   // B-operand strides: 80 % 64 == 16 -> lane halves hit disjoint banks
#define W2_LD   80
#define WC_LD   80

struct Params {
  const float* state;
  const float* wr_w1; const float* wr_b1; const float* wr_w2; const float* wr_b2;
  const float* wh_w1; const float* wh_b1; const float* wh_w2; const float* wh_b2;
  const float* rel_rh1; const float* brel_rh1; const float* root_rh1;
  const float* rel_hr1; const float* brel_hr1; const float* root_hr1;
  const float* rel_hh1; const float* brel_hh1; const float* root_hh1;
  const float* rel_hr2; const float* brel_hr2; const float* root_hr2;
  const float* vn_w1; const float* vn_b1; const float* vn_w2; const float* vn_b2;
  const float* vn_w3; const float* vn_b3;
  float* out;
};

// One 16x16 fp32 output tile via V_WMMA_F32_16X16X4_F32, K-chained in steps of 4.
// f32 A(16x4): lanes 0-15 hold M=0..15 K={0,1}; lanes 16-31 K={2,3}.
// f32 B(4x16): lanes 0-15 hold N=0..15 K rows {0,1}; lanes 16-31 rows {2,3}.
template<int K, int LDA, int LDB>
__device__ __forceinline__ v8f wmma_tile(const float* A, int rowBase,
                                         const float* Bm, int colBase,
                                         v8f acc, int lane) {
  const int hi = lane >> 4;
  const int lo = lane & 15;
  const int rowA = rowBase + lo;
  const int colB = colBase + lo;
#pragma unroll
  for (int kk = 0; kk < K; kk += 4) {
    const int k0 = kk + 2 * hi;
    v2f a, b;
    a.x = A[rowA * LDA + k0];
    a.y = A[rowA * LDA + k0 + 1];
    b.x = Bm[k0 * LDB + colB];
    b.y = Bm[(k0 + 1) * LDB + colB];
    acc = __builtin_amdgcn_wmma_f32_16x16x4_f32(false, a, false, b, (short)0,
                                                acc, false, false);
  }
  return acc;
}

__device__ __forceinline__ float relu(float x) { return x > 0.f ? x : 0.f; }

__global__ __launch_bounds__(256) void ValueNetwork_fused_kernel(Params p) {
  __shared__ float sWh1[8 * W1_LD];    // wh_w1 padded (K=8 rows)
  __shared__ float sWh2[64 * W2_LD];   // wh_w2
  __shared__ float sW1c[32 * WC_LD];   // root_rh1 + root_hh1 - rel_hh1
  __shared__ float sB1[64], sB2[32], sBd1[32];
  __shared__ float sHS[64 * HS_LD];
  __shared__ float sA1[64 * A1_LD];
  __shared__ float sH0[64 * H_LD];
  __shared__ float sH1[64 * H_LD];
  __shared__ float sSelf[8];
  __shared__ float sT1[64];
  __shared__ float sR0[32], sR1[32], sR2[32], sD1[32];
  __shared__ float sSum0[32], sSum1[32];
  __shared__ float sV1[152], sV2[104];
  __shared__ float sRed[32];

  const int tid  = threadIdx.x;
  const int lane = tid & 31;
  const int wv   = tid >> 5;

  // ---- One-time per-block weight staging (weights are L2-resident) ----
  for (int e = tid; e < 8 * 64; e += 256) {
    int r = e >> 6, c = e & 63;
    sWh1[r * W1_LD + c] = (r < 7) ? p.wh_w1[r * 64 + c] : 0.f;
  }
  for (int e = tid; e < 64 * 32; e += 256) {
    int r = e >> 5, c = e & 31;
    sWh2[r * W2_LD + c] = p.wh_w2[e];
  }
  for (int e = tid; e < 32 * 32; e += 256) {
    int r = e >> 5, c = e & 31;
    sW1c[r * WC_LD + c] = p.root_rh1[e] + p.root_hh1[e] - p.rel_hh1[e];
  }
  if (tid < 64) sB1[tid] = p.wh_b1[tid];
  if (tid < 32) {
    sB2[tid]  = p.wh_b2[tid];
    sBd1[tid] = p.brel_rh1[tid] + p.brel_hh1[tid];
  }
  __syncthreads();

  for (int b = blockIdx.x; b < NB; b += gridDim.x) {
    const float* srow = p.state + (size_t)b * NH * 13;

    // prefetch next batch item's state rows (L2 speculative)
    int bn = b + gridDim.x;
    if (bn < NB && tid * 16 < NH * 13)
      __builtin_prefetch(p.state + (size_t)bn * NH * 13 + tid * 16, 0, 1);

    // ---- Stage state -> LDS; zero column-sum accumulators ----
    for (int e = tid; e < NH * 8; e += 256) {
      int j = e >> 3, k = e & 7;
      sHS[j * HS_LD + k] = (k < 7) ? srow[j * 13 + 6 + k] : 0.f;
    }
    if (tid < 8) sSelf[tid] = (tid < 6) ? srow[tid] : 0.f;
    if (tid < 32) sSum0[tid] = 0.f;
    else if (tid < 64) sSum1[tid - 32] = 0.f;
    __syncthreads();

    // ---- GEMM1: (64x8) @ (8x64) -> a1 = relu(. + wh_b1). 16 tiles, 2/wave ----
    {
      const int hi = lane >> 4, lo = lane & 15;
#pragma unroll
      for (int ti = 0; ti < 2; ++ti) {
        const int t = wv * 2 + ti;
        const int mt = t >> 2, nt = t & 3;
        v8f acc = {0.f, 0.f, 0.f, 0.f, 0.f, 0.f, 0.f, 0.f};
        acc = wmma_tile<8, HS_LD, W1_LD>(sHS, mt * 16, sWh1, nt * 16, acc, lane);
        const int col = nt * 16 + lo;
        const float bias = sB1[col];
#pragma unroll
        for (int j = 0; j < 8; ++j) {
          int row = mt * 16 + j + 8 * hi;
          sA1[row * A1_LD + col] = relu(acc[j] + bias);
        }
      }
    }
    // self-path hidden (independent of a1)
    if (tid < 64) {
      float a = p.wr_b1[tid];
#pragma unroll
      for (int i = 0; i < 6; ++i) a += sSelf[i] * p.wr_w1[i * 64 + tid];
      sT1[tid] = relu(a);
    }
    __syncthreads();

    // ---- GEMM2: (64x64) @ (64x32) -> h0 = relu(. + wh_b2). 8 tiles, 1/wave ----
    {
      const int hi = lane >> 4, lo = lane & 15;
      const int mt = wv >> 1, nt = wv & 1;
      v8f acc = {0.f, 0.f, 0.f, 0.f, 0.f, 0.f, 0.f, 0.f};
      acc = wmma_tile<64, A1_LD, W2_LD>(sA1, mt * 16, sWh2, nt * 16, acc, lane);
      const int col = nt * 16 + lo;
      const float bias = sB2[col];
#pragma unroll
      for (int j = 0; j < 8; ++j) {
        int row = mt * 16 + j + 8 * hi;
        sH0[row * H_LD + col] = relu(acc[j] + bias);
      }
    }
    if (tid < 32) {  // r0
      float a = p.wr_b2[tid];
#pragma unroll 8
      for (int j = 0; j < 64; ++j) a += sT1[j] * p.wr_w2[j * 32 + tid];
      sR0[tid] = relu(a);
    }
    __syncthreads();

    // ---- sum0 = column sums of h0 (ds_add_f32 shared atomics) ----
    {
      float part = 0.f;
#pragma unroll
      for (int i = 0; i < 8; ++i) part += sH0[(wv * 8 + i) * H_LD + lane];
      atomicAdd(&sSum0[lane], part);
    }
    __syncthreads();

    // ---- r1 and the per-batch broadcast d1 (layer-1 vector algebra) ----
    if (tid < 32) {
      const int n = tid;
      float r1 = p.brel_hr1[n];
      float d1 = sBd1[n];
#pragma unroll 8
      for (int k = 0; k < 32; ++k) {
        const float s0 = sSum0[k], r0 = sR0[k];
        r1 += s0 * p.rel_hr1[k * 32 + n] + r0 * p.root_hr1[k * 32 + n];
        d1 += r0 * p.rel_rh1[k * 32 + n] + s0 * p.rel_hh1[k * 32 + n];
      }
      sR1[n] = relu(r1);
      sD1[n] = d1;
    }
    __syncthreads();

    // ---- GEMM3: h1 = relu(d1 + h0 @ (root_rh1+root_hh1-rel_hh1)) ----
    {
      const int hi = lane >> 4, lo = lane & 15;
      const int mt = wv >> 1, nt = wv & 1;
      v8f acc = {0.f, 0.f, 0.f, 0.f, 0.f, 0.f, 0.f, 0.f};
      acc = wmma_tile<32, H_LD, WC_LD>(sH0, mt * 16, sW1c, nt * 16, acc, lane);
      const int col = nt * 16 + lo;
      const float dd = sD1[col];
#pragma unroll
      for (int j = 0; j < 8; ++j) {
        int row = mt * 16 + j + 8 * hi;
        sH1[row * H_LD + col] = relu(acc[j] + dd);
      }
    }
    __syncthreads();

    // ---- sum1; note: layer-2 new_h is dead code (value head uses r only) ----
    {
      float part = 0.f;
#pragma unroll
      for (int i = 0; i < 8; ++i) part += sH1[(wv * 8 + i) * H_LD + lane];
      atomicAdd(&sSum1[lane], part);
    }
    __syncthreads();

    if (tid < 32) {  // r2 = relu(sum1@rel_hr2 + brel_hr2 + r1@root_hr2)
      const int n = tid;
      float r2 = p.brel_hr2[n];
#pragma unroll 8
      for (int k = 0; k < 32; ++k)
        r2 += sSum1[k] * p.rel_hr2[k * 32 + n] + sR1[k] * p.root_hr2[k * 32 + n];
      sR2[n] = relu(r2);
    }
    __syncthreads();

    // ---- Value head: 32 -> 150 -> 100 -> 1 (weights stay hot in L2) ----
    if (tid < 150) {
      float a = p.vn_b1[tid];
#pragma unroll 8
      for (int k = 0; k < 32; ++k) a += sR2[k] * p.vn_w1[k * 150 + tid];
      sV1[tid] = relu(a);
    }
    __syncthreads();
    if (tid < 100) {
      float a = p.vn_b2[tid];
      for (int k = 0; k < 150; ++k) a += sV1[k] * p.vn_w2[k * 100 + tid];
      sV2[tid] = relu(a);
    }
    __syncthreads();
    if (tid < 32) {
      float part = sV2[tid] * p.vn_w3[tid]
                 + sV2[tid + 32] * p.vn_w3[tid + 32]
                 + sV2[tid + 64] * p.vn_w3[tid + 64];
      if (tid < 4) part += sV2[tid + 96] * p.vn_w3[tid + 96];
      sRed[tid] = part;
    }
    __syncthreads();
    if (tid == 0) {
      float o = p.vn_b3[0];
#pragma unroll
      for (int i = 0; i < 32; ++i) o += sRed[i];
      p.out[b] = o;
    }
    __syncthreads();  // protect LDS reuse for next batch item
  }
}

extern "C" void kernel_launch(void* const* d_in, const int* in_sizes, int n_in,
                              void* d_out, int out_size, void* d_ws, size_t ws_size,
                              hipStream_t stream) {
  (void)in_sizes; (void)n_in; (void)out_size; (void)d_ws; (void)ws_size;
  Params p;
  p.state    = (const float*)d_in[0];
  // d_in[1] = dropout (unused, == 0)
  p.wr_w1    = (const float*)d_in[2];
  p.wr_b1    = (const float*)d_in[3];
  p.wr_w2    = (const float*)d_in[4];
  p.wr_b2    = (const float*)d_in[5];
  p.wh_w1    = (const float*)d_in[6];
  p.wh_b1    = (const float*)d_in[7];
  p.wh_w2    = (const float*)d_in[8];
  p.wh_b2    = (const float*)d_in[9];
  p.rel_rh1  = (const float*)d_in[10];
  p.brel_rh1 = (const float*)d_in[11];
  p.root_rh1 = (const float*)d_in[12];
  p.rel_hr1  = (const float*)d_in[13];
  p.brel_hr1 = (const float*)d_in[14];
  p.root_hr1 = (const float*)d_in[15];
  p.rel_hh1  = (const float*)d_in[16];
  p.brel_hh1 = (const float*)d_in[17];
  p.root_hh1 = (const float*)d_in[18];
  // layer-2: only hr path is live (new_h of layer 2 is dead)
  p.rel_hr2  = (const float*)d_in[22];
  p.brel_hr2 = (const float*)d_in[23];
  p.root_hr2 = (const float*)d_in[24];
  p.vn_w1    = (const float*)d_in[28];
  p.vn_b1    = (const float*)d_in[29];
  p.vn_w2    = (const float*)d_in[30];
  p.vn_b2    = (const float*)d_in[31];
  p.vn_w3    = (const float*)d_in[32];
  p.vn_b3    = (const float*)d_in[33];
  p.out      = (float*)d_out;

  dim3 grid(2048), block(256);
  hipLaunchKernelGGL(ValueNetwork_fused_kernel, grid, block, 0, stream, p);
}